// NeuralCA_3444563771649
// MI455X (gfx1250) — compile-verified
//
#include <hip/hip_runtime.h>

// NeuralCA step: N = conv3x3(x, 16 kernels); cond = sigmoid(10*(N - r));
// out = clip(x + sum_k cond_k * (p_k - x), 0, 1)
//
// Mapping: one wave32 = one tile of 16 consecutive pixels along W.
//   WMMA f32 16x16x4:  A = [16 pixels x 4 taps] (im2col), B = [4 taps x 16 kernels],
//   C accum over 3 WMMAs covering the 9 taps (K padded to 12).
// C/D layout: lane L owns kernel n = L%16 for 8 pixel-rows -> per-lane sigmoid/update,
// cross-kernel sum = 16-lane butterfly reduction.

#define HH 512
#define WW 512
#define GAIN 10.0f

typedef __attribute__((ext_vector_type(2))) float v2f;
typedef __attribute__((ext_vector_type(8))) float v8f;

__global__ __launch_bounds__(256) void nca_step_kernel(
    const float* __restrict__ x,      // [16,1,512,512]
    const float* __restrict__ ker,    // [16,1,3,3]
    const float* __restrict__ rea,    // [16]
    const float* __restrict__ pro,    // [16]
    float* __restrict__ out)          // [16,1,512,512]
{
    const int lane = threadIdx.x & 31;
    const int wave = threadIdx.x >> 5;
    const int m  = lane & 15;   // A: pixel-in-tile; B/C: kernel index n
    const int kg = lane >> 4;   // K-group: 0 -> K{0,1}, 1 -> K{2,3}

    const int tile = blockIdx.x * 8 + wave;      // 262144 tiles total
    const int x0   = (tile & 31) << 4;           // 32 tiles per image row
    const int row  = tile >> 5;
    const int y    = row & (HH - 1);
    const int b    = row >> 9;                   // row / 512

    const size_t plane = (size_t)b * HH;

    // ---- B matrix: conv weights for this lane's kernel n=m at taps K=2kg,2kg+1 ----
    const int k0 = 2 * kg;
    v2f bw0 = { ker[m * 9 + k0],     ker[m * 9 + k0 + 1] };
    v2f bw1 = { ker[m * 9 + 4 + k0], ker[m * 9 + 5 + k0] };
    v2f bw2 = { (kg == 0) ? ker[m * 9 + 8] : 0.0f, 0.0f };

    // ---- A matrix: im2col loads (zero-padded borders), pixel = x0 + m ----
    auto ld = [&](int t) -> float {
        int dy = t / 3 - 1, dx = t % 3 - 1;
        int yy = y + dy, xx = x0 + m + dx;
        if ((unsigned)yy >= (unsigned)HH || (unsigned)xx >= (unsigned)WW) return 0.0f;
        return x[(plane + (size_t)yy) * WW + xx];
    };
    v2f a0 = { ld(k0),     ld(k0 + 1) };
    v2f a1 = { ld(4 + k0), ld(5 + k0) };
    v2f a2 = { (kg == 0) ? ld(8) : 0.0f, 0.0f };

    // center x for this lane's pixel (tap dy=0,dx=0; cache hit)
    const float xc = x[(plane + (size_t)y) * WW + x0 + m];

    // speculative prefetch of the next image row for this tile (gfx1250 global_prefetch_b8)
    if (y + 1 < HH) __builtin_prefetch(&x[(plane + (size_t)(y + 1)) * WW + x0 + m], 0, 0);

    // ---- conv via 3x V_WMMA_F32_16X16X4_F32 (K = 9 taps padded to 12) ----
    v8f acc = {0.f, 0.f, 0.f, 0.f, 0.f, 0.f, 0.f, 0.f};
    acc = __builtin_amdgcn_wmma_f32_16x16x4_f32(false, a0, false, bw0, (short)0, acc, false, false);
    acc = __builtin_amdgcn_wmma_f32_16x16x4_f32(false, a1, false, bw1, (short)0, acc, false, false);
    acc = __builtin_amdgcn_wmma_f32_16x16x4_f32(false, a2, false, bw2, (short)0, acc, false, false);

    // ---- per-(pixel,kernel) activation, then sum over the 16 kernels ----
    const float r = rea[m];
    const float p = pro[m];

    float dval[8];
#pragma unroll
    for (int j = 0; j < 8; ++j) {
        // C/D row j: lanes 0-15 -> pixel j, lanes 16-31 -> pixel j+8
        float z    = GAIN * (acc[j] - r);
        // sigmoid(z) = 1 / (1 + 2^(-z*log2(e)))
        float e    = __builtin_amdgcn_exp2f(z * -1.44269504088896340736f);
        float cond = __builtin_amdgcn_rcpf(1.0f + e);
        float xcj  = __shfl(xc, j + 8 * kg, 32);   // broadcast center x of this row's pixel
        float upd  = cond * (p - xcj);
        // butterfly sum over the 16 kernels (stays within each 16-lane half)
        upd += __shfl_xor(upd, 1, 32);
        upd += __shfl_xor(upd, 2, 32);
        upd += __shfl_xor(upd, 4, 32);
        upd += __shfl_xor(upd, 8, 32);
        dval[j] = fminf(fmaxf(xcj + upd, 0.0f), 1.0f);
    }

    // lanes m<8 of each half write pixel (m + 8*kg) using row j=m
    float sel = dval[0];
#pragma unroll
    for (int j = 1; j < 8; ++j) sel = (m == j) ? dval[j] : sel;
    if (m < 8)
        out[(plane + (size_t)y) * WW + x0 + m + 8 * kg] = sel;
}

extern "C" void kernel_launch(void* const* d_in, const int* in_sizes, int n_in,
                              void* d_out, int out_size, void* d_ws, size_t ws_size,
                              hipStream_t stream) {
    const float* x   = (const float*)d_in[0];  // 16*1*512*512
    const float* ker = (const float*)d_in[1];  // 16*1*3*3
    const float* rea = (const float*)d_in[2];  // 16
    const float* pro = (const float*)d_in[3];  // 16
    float* out = (float*)d_out;                // 16*1*512*512

    // tiles = 16*512*(512/16) = 262144 ; 8 waves (tiles) per 256-thread block
    dim3 grid(262144 / 8), block(256);
    hipLaunchKernelGGL(nca_step_kernel, grid, block, 0, stream, x, ker, rea, pro, out);
}